// GCN_78589311582712
// MI455X (gfx1250) — compile-verified
//
#include <hip/hip_runtime.h>

typedef float v2f __attribute__((ext_vector_type(2)));
typedef float v8f __attribute__((ext_vector_type(8)));

// ---------------------------------------------------------------- utilities
__global__ void zero_kernel(float* __restrict__ p, long n) {
  long i = (long)blockIdx.x * blockDim.x + threadIdx.x;
  long stride = (long)gridDim.x * blockDim.x;
  for (; i < n; i += stride) p[i] = 0.0f;
}

__global__ void degree_kernel(const int* __restrict__ src, const int* __restrict__ dst,
                              float* __restrict__ deg_src, float* __restrict__ deg_dst,
                              long E) {
  long i = (long)blockIdx.x * blockDim.x + threadIdx.x;
  if (i < E) {
    unsafeAtomicAdd(&deg_src[src[i]], 1.0f);
    unsafeAtomicAdd(&deg_dst[dst[i]], 1.0f);
  }
}

__global__ void norm_kernel(const float* __restrict__ deg_src, const float* __restrict__ deg_dst,
                            float* __restrict__ norm_src, float* __restrict__ norm_dst, int n) {
  int i = blockIdx.x * blockDim.x + threadIdx.x;
  if (i < n) {
    float a = deg_src[i], b = deg_dst[i];
    norm_src[i] = (a > 0.0f) ? __frsqrt_rn(fmaxf(a, 1.0f)) : 0.0f;
    norm_dst[i] = (b > 0.0f) ? __frsqrt_rn(fmaxf(b, 1.0f)) : 0.0f;
  }
}

// ---------------------------------------------------------------- WMMA GEMM
// out[N, Nout] = (in[N,128] @ W[128,Nout]) * scale[row]? + bias[col]?
// grid.x = N/16, block = (32, COLS/16). One wave per 16x16 output tile.
//
// A tile (16x128, row-major) is staged into LDS with the gfx1250 async DMA
// path: global_load_async_to_lds_b128 (ASYNCcnt-tracked), since global and
// LDS layouts are byte-identical. W is staged via registers because it is
// swizzled into (k/2, n) float2 pairs and zero-padded past Nout, so the
// inner loop is exactly: ds_load_b64 (A) + ds_load_b64 (B) + v_wmma.
// No guards inside the K loop -> EXEC stays all-ones for every WMMA.
template <int COLS>
__global__ void wmma_gemm_kernel(const float* __restrict__ in,
                                 const float* __restrict__ W,
                                 const float* __restrict__ scale,   // nullable, per-row
                                 const float* __restrict__ bias,    // nullable, per-col
                                 float* __restrict__ out,
                                 int Ntotal, int Nout) {
  constexpr int K = 128;
  __shared__ float  As[16 * K];            // 8 KB
  __shared__ float2 Bs[(K / 2) * COLS];    // COLS=128: 64 KB; COLS=48: 24 KB

  const int rowBase = blockIdx.x * 16;
  const int tid  = threadIdx.y * 32 + threadIdx.x;
  const int nthr = blockDim.y * 32;

  // ---- stage A tile via async global->LDS DMA (16-byte chunks, row-clamped)
  {
    const unsigned ldsBase = (unsigned)(size_t)(void*)As;
    for (int i = tid; i < (16 * K) / 4; i += nthr) {   // 512 x 16B chunks
      int r = rowBase + (i >> 5);                      // 32 chunks per row
      if (r >= Ntotal) r = Ntotal - 1;
      const int c = (i & 31) * 4;
      unsigned lds_off = ldsBase + (unsigned)i * 16u;
      unsigned long long ga =
          (unsigned long long)(size_t)(in + (long)r * K + c);
      asm volatile("global_load_async_to_lds_b128 %0, %1, off"
                   :: "v"(lds_off), "v"(ga) : "memory");
    }
  }

  // ---- stage W swizzled + zero-padded: Bs[kp*COLS + n] = {W[2kp][n], W[2kp+1][n]}
  for (int i = tid; i < (K / 2) * COLS; i += nthr) {
    int kp = i / COLS;
    int n  = i - kp * COLS;
    float2 w = make_float2(0.0f, 0.0f);
    if (n < Nout) {
      w.x = W[(long)(2 * kp) * Nout + n];
      w.y = W[(long)(2 * kp + 1) * Nout + n];
    }
    Bs[i] = w;
  }

  asm volatile("s_wait_asynccnt 0x0" ::: "memory");   // A-tile DMA complete
  __syncthreads();

  const int lane = threadIdx.x;               // 0..31, wave32
  const int m    = lane & 15;                 // A row / B col within tile
  const int koff = (lane >> 4) * 2;           // lanes 16..31 hold K+2,K+3
  const int n    = threadIdx.y * 16 + m;      // output column (within COLS)

  const float2* As2 = (const float2*)As;
  const int aRowBase = m * (K / 2);

  v8f acc = {};
#pragma unroll
  for (int k = 0; k < K; k += 4) {
    const int kp = (k + koff) >> 1;           // even: (k+koff)/2
    float2 af = As2[aRowBase + kp];
    float2 bf = Bs[kp * COLS + n];
    v2f a, b;
    a.x = af.x; a.y = af.y;
    b.x = bf.x; b.y = bf.y;
    acc = __builtin_amdgcn_wmma_f32_16x16x4_f32(
        /*neg_a=*/false, a, /*neg_b=*/false, b,
        /*c_mod=*/(short)0, acc, /*reuse_a=*/false, /*reuse_b=*/false);
  }

  // C/D layout: VGPR r -> M = r + (lane<16 ? 0 : 8), N = lane&15
  const int rBase = rowBase + ((lane < 16) ? 0 : 8);
  const bool nok = (n < Nout);
  const float badd = (bias != nullptr && nok) ? bias[n] : 0.0f;
#pragma unroll
  for (int r = 0; r < 8; ++r) {
    int rr = rBase + r;
    if (rr < Ntotal && nok) {
      float v = acc[r];
      if (scale != nullptr) v *= scale[rr];
      v += badd;
      out[(long)rr * Nout + n] = v;
    }
  }
}

// ---------------------------------------------------------------- edge scatter
// One wave per edge: gather msg[src[e]] (float4 per lane = 128 floats),
// scatter-add into agg[dst[e]] via fp32 global atomics (no-return -> STOREcnt).
// msg/agg are ~51 MB each -> L2-resident (192 MB), so the random traffic
// stays on-chip; HBM only sees the streaming index reads.
__global__ void scatter_kernel(const float* __restrict__ msg,
                               const int* __restrict__ src,
                               const int* __restrict__ dst,
                               float* __restrict__ agg, long E) {
  const int wavesPerBlock = blockDim.x >> 5;
  long e = (long)blockIdx.x * wavesPerBlock + (threadIdx.x >> 5);
  const int lane = threadIdx.x & 31;
  if (e >= E) return;

  // stream-prefetch the index arrays ahead (speculative, gfx1250 global_prefetch_b8)
  __builtin_prefetch(&src[e + 2048], 0, 0);
  __builtin_prefetch(&dst[e + 2048], 0, 0);

  const int s = src[e];
  const int d = dst[e];
  const float4 v = ((const float4*)(msg + (long)s * 128))[lane];
  float* ap = agg + (long)d * 128 + lane * 4;
  unsafeAtomicAdd(ap + 0, v.x);
  unsafeAtomicAdd(ap + 1, v.y);
  unsafeAtomicAdd(ap + 2, v.z);
  unsafeAtomicAdd(ap + 3, v.w);
}

// ---------------------------------------------------------------- epilogue
// agg = relu(agg * norm_dst[row] + bias[col]), in place
__global__ void finalize_kernel(float* __restrict__ agg,
                                const float* __restrict__ norm_dst,
                                const float* __restrict__ bias, long total) {
  long i = (long)blockIdx.x * blockDim.x + threadIdx.x;
  if (i < total) {
    int row = (int)(i >> 7);
    int col = (int)(i & 127);
    float v = agg[i] * norm_dst[row] + bias[col];
    agg[i] = fmaxf(v, 0.0f);
  }
}

// ---------------------------------------------------------------- launcher
extern "C" void kernel_launch(void* const* d_in, const int* in_sizes, int n_in,
                              void* d_out, int out_size, void* d_ws, size_t ws_size,
                              hipStream_t stream) {
  const float* x   = (const float*)d_in[0];
  const int*   src = (const int*)d_in[1];
  const int*   dst = (const int*)d_in[2];
  const float* W1  = (const float*)d_in[3];
  const float* b1  = (const float*)d_in[4];
  const float* W2  = (const float*)d_in[5];
  const float* b2  = (const float*)d_in[6];
  const float* W3  = (const float*)d_in[7];
  const float* b3  = (const float*)d_in[8];
  const float* Wc  = (const float*)d_in[9];
  const float* bc  = (const float*)d_in[10];

  const int  N = in_sizes[0] / 128;   // 100000
  const long E = in_sizes[1];         // 1600000
  const int  C = in_sizes[10];        // 40

  float* ws       = (float*)d_ws;
  float* norm_src = ws;
  float* norm_dst = ws + (long)N;
  float* deg_src  = ws + 2L * N;
  float* deg_dst  = ws + 3L * N;      // contiguous with deg_src
  float* msg      = ws + 4L * N;
  float* agg      = msg + (long)N * 128;

  // ---- degree norms
  zero_kernel<<<512, 256, 0, stream>>>(deg_src, 2L * N);
  degree_kernel<<<(int)((E + 255) / 256), 256, 0, stream>>>(src, dst, deg_src, deg_dst, E);
  norm_kernel<<<(N + 255) / 256, 256, 0, stream>>>(deg_src, deg_dst, norm_src, norm_dst, N);

  const int rowTiles = (N + 15) / 16;           // 6250
  const float* Wl[3] = {W1, W2, W3};
  const float* bl[3] = {b1, b2, b3};
  const float* layer_in = x;

  for (int l = 0; l < 3; ++l) {
    // msg = (layer_in @ W) * norm_src[row]
    wmma_gemm_kernel<128><<<rowTiles, dim3(32, 8), 0, stream>>>(
        layer_in, Wl[l], norm_src, nullptr, msg, N, 128);
    // agg = 0
    zero_kernel<<<2048, 256, 0, stream>>>(agg, (long)N * 128);
    // agg[dst] += msg[src]
    scatter_kernel<<<(int)((E + 7) / 8), 256, 0, stream>>>(msg, src, dst, agg, E);
    // agg = relu(agg * norm_dst + b)
    finalize_kernel<<<(int)(((long)N * 128 + 255) / 256), 256, 0, stream>>>(
        agg, norm_dst, bl[l], (long)N * 128);
    layer_in = agg;
  }

  // classifier: out = agg @ Wc + bc   (Nout = 40 -> 48 padded cols, 3 waves)
  wmma_gemm_kernel<48><<<rowTiles, dim3(32, 3), 0, stream>>>(
      agg, Wc, nullptr, bc, (float*)d_out, N, C);
}